// Loss_42210938585684
// MI455X (gfx1250) — compile-verified
//
#include <hip/hip_runtime.h>

typedef __attribute__((ext_vector_type(2))) float v2f;
typedef __attribute__((ext_vector_type(8))) float v8f;

// ---------------------------------------------------------------------------
// IOU of two corner-form boxes, matching the reference (eps = 1e-7).
// ---------------------------------------------------------------------------
__device__ __forceinline__ float iou_corners(float ax1, float ay1, float ax2, float ay2,
                                             float bx1, float by1, float bx2, float by2) {
    float xl = fmaxf(ax1, bx1);
    float yt = fmaxf(ay1, by1);
    float xr = fminf(ax2, bx2);
    float yb = fminf(ay2, by2);
    float ix = fmaxf(xr - xl, 0.0f);
    float iy = fmaxf(yb - yt, 0.0f);
    float inter = ix * iy;
    float aa = fabsf((ax2 - ax1) * (ay2 - ay1));
    float ba = fabsf((bx2 - bx1) * (by2 - by1));
    return inter / (aa + ba - inter + 1e-7f);
}

// ---------------------------------------------------------------------------
// Wave32 sum via V_WMMA_F32_16X16X4_F32.
// A (16x4 f32): lane L, VGPR0/1. We place each lane's partial in A[.,.] and
// zero the second slot -> A holds exactly the 32 partials. B = all-ones 4x16,
// so D[m,n] = sum_k A[m,k] (exact f32 FMA chain). Each lane sums its 8 C/D
// VGPRs: lanes 0-15 then hold sum over M=0..7, lanes 16-31 sum over M=8..15.
// One shfl_down(16) gives the full wave sum on lane 0.
// Requires EXEC == all 1s (callers keep every lane alive).
// ---------------------------------------------------------------------------
__device__ __forceinline__ float wave_sum_wmma(float s) {
    v2f a; a[0] = s;    a[1] = 0.0f;
    v2f b; b[0] = 1.0f; b[1] = 1.0f;
    v8f c = {};
    c = __builtin_amdgcn_wmma_f32_16x16x4_f32(false, a, false, b, (short)0, c, false, false);
    float r = c[0] + c[1] + c[2] + c[3] + c[4] + c[5] + c[6] + c[7];
    r += __shfl_down(r, 16, 32);
    return r;  // valid on lane 0
}

// Deterministic block reduction: WMMA wave sums -> LDS -> serial sum by t0.
__device__ __forceinline__ void block_reduce_store(float s, float* dst, float scale) {
    __shared__ float swave[32];
    const int lane = threadIdx.x & 31;
    const int wv   = threadIdx.x >> 5;
    float r = wave_sum_wmma(s);
    if (lane == 0) swave[wv] = r;
    __syncthreads();
    if (threadIdx.x == 0) {
        float tot = 0.0f;
        const int nw = (int)(blockDim.x >> 5);
        for (int w = 0; w < nw; ++w) tot += swave[w];
        dst[0] = tot * scale;
    }
}

// ---------------------------------------------------------------------------
// Phase 1: one thread per grid cell; per-block partial sums into d_ws.
// Layout: [N, 7, 7, 30] row-major; lt[i][j] = (j/7, i/7).
// ---------------------------------------------------------------------------
__global__ void yolo_loss_partial(const float* __restrict__ p_all,
                                  const float* __restrict__ t_all,
                                  float* __restrict__ partial,
                                  int total_cells) {
    const int idx = blockIdx.x * blockDim.x + threadIdx.x;
    float s = 0.0f;
    if (idx < total_cells) {   // predicate body only; all lanes stay alive for WMMA
        const float inv_s = 1.0f / 7.0f;
        const int cell = idx % 49;
        const float ltx = (float)(cell % 7) * inv_s;
        const float lty = (float)(cell / 7) * inv_s;

        const float2* __restrict__ p2 = (const float2*)(p_all + (long long)idx * 30);
        const float2* __restrict__ t2 = (const float2*)(t_all + (long long)idx * 30);
        float pv[30], tv[30];
#pragma unroll
        for (int k = 0; k < 15; ++k) {
            float2 vp = p2[k]; pv[2 * k] = vp.x; pv[2 * k + 1] = vp.y;
            float2 vt = t2[k]; tv[2 * k] = vt.x; tv[2 * k + 1] = vt.y;
        }

        const float coord = (tv[4] > 0.0f)  ? 1.0f : 0.0f;
        const float noobj = (tv[4] == 0.0f) ? 1.0f : 0.0f;

        // predicted box 1: center-offset/size parametrization
        const float b1x1 = ltx + pv[0] * inv_s - pv[2] * 0.5f;
        const float b1y1 = lty + pv[1] * inv_s - pv[3] * 0.5f;
        const float b1x2 = pv[2] + b1x1;
        const float b1y2 = pv[3] + b1y1;

        // predicted box 2 (faithful to source: p[5:7] * (1/S - 0.5))
        const float c2 = inv_s - 0.5f;
        const float b2x1 = ltx + pv[5] * c2;
        const float b2y1 = lty + pv[6] * c2;
        const float b2x2 = pv[7] + b2x1;
        const float b2y2 = pv[8] + b2y1;

        // target box
        const float tx1 = ltx + tv[0] * inv_s - tv[2] * 0.5f;
        const float ty1 = lty + tv[1] * inv_s - tv[3] * 0.5f;
        const float tx2 = tv[2] + tx1;
        const float ty2 = tv[3] + ty1;

        const float iou1 = iou_corners(b1x1, b1y1, b1x2, b1y2, tx1, ty1, tx2, ty2);
        const float iou2 = iou_corners(b2x1, b2y1, b2x2, b2y2, tx1, ty1, tx2, ty2);
        const bool sel = (iou1 >= iou2);

        const float bx = sel ? pv[0] : pv[5];
        const float by = sel ? pv[1] : pv[6];
        const float bw = sel ? pv[2] : pv[7];
        const float bh = sel ? pv[3] : pv[8];
        const float bc = sel ? pv[4] : pv[9];

        const float dx = bx - tv[0], dy = by - tv[1];
        const float xy_se = dx * dx + dy * dy;

        const float whw = copysignf(sqrtf(fabsf(bw)), bw);
        const float whh = copysignf(sqrtf(fabsf(bh)), bh);
        const float dw = whw - sqrtf(tv[2]);
        const float dh = whh - sqrtf(tv[3]);
        const float wh_se = dw * dw + dh * dh;

        const float dob = fmaxf(iou1, iou2) - bc;
        const float obj_se = dob * dob;

        const float dn1 = pv[4] - tv[4], dn2 = pv[9] - tv[9];
        const float noobj_se = dn1 * dn1 + dn2 * dn2;

        float label_se = 0.0f;
#pragma unroll
        for (int k = 10; k < 30; ++k) {
            const float d = pv[k] - tv[k];
            label_se += d * d;
        }

        s = coord * (5.0f * (xy_se + wh_se) + obj_se + label_se) + 0.5f * noobj * noobj_se;
    }

    block_reduce_store(s, partial + blockIdx.x, 1.0f);
}

// ---------------------------------------------------------------------------
// Phase 2: single block folds the per-block partials (fixed order ->
// deterministic), scales by 1/N, writes the scalar loss.
// ---------------------------------------------------------------------------
__global__ void yolo_loss_final(const float* __restrict__ partial, int nparts,
                                float* __restrict__ out, float inv_n) {
    float s = 0.0f;
    for (int k = threadIdx.x; k < nparts; k += blockDim.x) s += partial[k];
    block_reduce_store(s, out, inv_n);
}

extern "C" void kernel_launch(void* const* d_in, const int* in_sizes, int n_in,
                              void* d_out, int out_size, void* d_ws, size_t ws_size,
                              hipStream_t stream) {
    const float* modely  = (const float*)d_in[0];
    const float* targety = (const float*)d_in[1];

    const int total_cells = in_sizes[0] / 30;  // N * 7 * 7
    const int n_img       = total_cells / 49;  // N

    const int threads = 256;
    const int blocks  = (total_cells + threads - 1) / threads;  // 1568 for N=8192

    float* partial = (float*)d_ws;  // blocks * 4 bytes (~6.3 KB)

    yolo_loss_partial<<<blocks, threads, 0, stream>>>(modely, targety, partial, total_cells);
    yolo_loss_final<<<1, threads, 0, stream>>>(partial, blocks, (float*)d_out,
                                               1.0f / (float)n_img);
}